// GravitationalPotential_26070451487123
// MI455X (gfx1250) — compile-verified
//
#include <hip/hip_runtime.h>

typedef __attribute__((ext_vector_type(2))) float v2f;
typedef __attribute__((ext_vector_type(8))) float v8f;

#define GCONST 0.001f
#define EPSCONST 1e-6f

// One 16x16 interaction tile. The WMMA computes d2 entirely:
//   A (16x4) = [-2x_i, -2y_i, -2z_i, 1]   (K3 slot = 1)
//   B (4x16) = [ x_j,   y_j,   z_j, nj]   (K3 row carries |qj|^2)
//   C        = niv[r] = |qi|^2 + EPS      (loop-invariant registers)
//   D[r]     = -2 qi.qj + nj + ni + eps = |qi-qj|^2 + eps
// MASKED=true additionally zeroes the i==j diagonal (only when J==I).
template <bool MASKED>
__device__ __forceinline__ float tile_sum(const v2f& afrag, const v8f& niv,
                                          const float* __restrict__ mi,
                                          int ibase,
                                          const float* __restrict__ qb,
                                          const float* __restrict__ m,
                                          int J, int hi, int lr) {
  const int col = J * 16 + lr;
  const float cx = qb[3 * col + 0];
  const float cy = qb[3 * col + 1];
  const float cz = qb[3 * col + 2];
  const float nj = cx * cx + cy * cy + cz * cz;
  const float mj = m[col];
  // lanes 0-15: VGPR0/1 = rows K0/K1 (x_j, y_j)
  // lanes 16-31: VGPR0/1 = rows K2/K3 (z_j, nj)
  v2f bfrag;
  bfrag.x = hi ? cz : cx;
  bfrag.y = hi ? nj : cy;

  const v8f d2 = __builtin_amdgcn_wmma_f32_16x16x4_f32(
      /*neg_a=*/false, afrag, /*neg_b=*/false, bfrag,
      /*c_mod=*/(short)0, niv, /*reuse_a=*/false, /*reuse_b=*/false);

  // sum_r mi*mj*rsqrt(d2) = mj * sum_r mi*rsqrt(d2)
  float t = 0.0f;
#pragma unroll
  for (int r = 0; r < 8; ++r) {
    float v = mi[r] * __builtin_amdgcn_rsqf(d2[r]);  // v_rsq_f32
    if (MASKED) v = ((ibase + r) == col) ? 0.0f : v;
    t += v;
  }
  return t * mj;
}

// Upper-triangle sweep of one row tile, split between two waves by parity.
// Diagonal tile (J==I) handled once (parity 0) with masking + 0.5 weight.
__device__ __forceinline__ float row_tile_sum(const float* __restrict__ qb,
                                              const float* __restrict__ m,
                                              int Itile, int parity,
                                              int tilesPerBatch, int hi,
                                              int lr) {
  // A fragment: lanes 0-15 = (-2x, -2y); lanes 16-31 = (-2z, 1)
  const int row = Itile * 16 + lr;
  const float rx = qb[3 * row + 0];
  const float ry = qb[3 * row + 1];
  const float rz = qb[3 * row + 2];
  v2f afrag;
  afrag.x = -2.0f * (hi ? rz : rx);
  afrag.y = hi ? 1.0f : (-2.0f * ry);

  // Loop-invariant C operand (row norms + EPS) and row masses
  const int ibase = Itile * 16 + 8 * hi;
  v8f niv;
  float mi[8];
#pragma unroll
  for (int r = 0; r < 8; ++r) {
    const int i = ibase + r;
    const float x = qb[3 * i + 0];
    const float y = qb[3 * i + 1];
    const float z = qb[3 * i + 2];
    niv[r] = x * x + y * y + z * z + EPSCONST;
    mi[r] = m[i];
  }

  float acc = 0.0f;
  if (parity == 0)  // wave-uniform (SGPR) -> scalar branch, EXEC untouched
    acc += 0.5f * tile_sum<true>(afrag, niv, mi, ibase, qb, m, Itile, hi, lr);
#pragma unroll 2
  for (int J = Itile + 2 - parity; J < tilesPerBatch; J += 2)
    acc += tile_sum<false>(afrag, niv, mi, ibase, qb, m, J, hi, lr);
  return acc;
}

// Wave w: batch b = w/256, slot s = w%256; pair p = s>>1 owns row tiles
// {p, 255-p} (balanced triangle halves), parity s&1 takes alternate J tiles.
__global__ __launch_bounds__(256) void grav_tiles_kernel(
    const float* __restrict__ q,   // (B, N, 3)
    const float* __restrict__ m,   // (N,)
    float* __restrict__ partial,   // (B * tilesPerBatch,)
    int N, int tilesPerBatch, int totalWaves) {
  // Wave-uniform control values forced into SGPRs -> scalar branches only.
  const int wave = __builtin_amdgcn_readfirstlane(
      blockIdx.x * (blockDim.x >> 5) + (threadIdx.x >> 5));
  if (wave >= totalWaves) return;  // scalar branch

  const int b      = wave / tilesPerBatch;
  const int s      = wave % tilesPerBatch;
  const int pair   = s >> 1;
  const int parity = s & 1;
  const int I1     = pair;
  const int I2     = tilesPerBatch - 1 - pair;

  const int lane = threadIdx.x & 31;
  const int hi   = lane >> 4;
  const int lr   = lane & 15;

  const float* qb = q + (size_t)b * N * 3;

  float acc = row_tile_sum(qb, m, I1, parity, tilesPerBatch, hi, lr) +
              row_tile_sum(qb, m, I2, parity, tilesPerBatch, hi, lr);

  // wave32 butterfly reduction
#pragma unroll
  for (int off = 16; off >= 1; off >>= 1)
    acc += __shfl_xor(acc, off, 32);

  if (lane == 0) partial[wave] = acc * (-GCONST);
}

// Deterministic per-batch tree reduction. Triangle sum already includes the
// /2 (off-diag tiles counted once; diag tiles pre-weighted by 0.5).
__global__ __launch_bounds__(256) void grav_reduce_kernel(
    const float* __restrict__ partial, float* __restrict__ out,
    int wavesPerBatch) {
  __shared__ float sdata[256];
  const int b = blockIdx.x;
  float sum = 0.0f;
  for (int t = threadIdx.x; t < wavesPerBatch; t += 256)
    sum += partial[(size_t)b * wavesPerBatch + t];
  sdata[threadIdx.x] = sum;
  __syncthreads();
#pragma unroll
  for (int off = 128; off >= 1; off >>= 1) {
    if ((int)threadIdx.x < off) sdata[threadIdx.x] += sdata[threadIdx.x + off];
    __syncthreads();
  }
  if (threadIdx.x == 0) out[b] = sdata[0];
}

extern "C" void kernel_launch(void* const* d_in, const int* in_sizes, int n_in,
                              void* d_out, int out_size, void* d_ws, size_t ws_size,
                              hipStream_t stream) {
  const float* q = (const float*)d_in[0];   // (B, N, 3) f32
  const float* m = (const float*)d_in[1];   // (N,) f32

  const int N = in_sizes[1];                // 4096
  const int B = in_sizes[0] / (N * 3);      // 8
  const int tilesPerBatch = N / 16;         // 256 (even: pairing I <-> 255-I)
  const int totalWaves = B * tilesPerBatch; // 2048 waves, ~128.5 tiles each

  float* partial = (float*)d_ws;            // totalWaves * 4 bytes = 8 KB

  const int wavesPerBlock = 8;              // 256 threads = 8 wave32
  const int blocks = (totalWaves + wavesPerBlock - 1) / wavesPerBlock;

  grav_tiles_kernel<<<blocks, 256, 0, stream>>>(q, m, partial, N,
                                                tilesPerBatch, totalWaves);
  grav_reduce_kernel<<<B, 256, 0, stream>>>(partial, (float*)d_out,
                                            tilesPerBatch);
}